// GATLayer_11381663334531
// MI455X (gfx1250) — compile-verified
//
#include <hip/hip_runtime.h>
#include <hip/hip_bf16.h>

typedef __attribute__((ext_vector_type(2))) float v2f;
typedef __attribute__((ext_vector_type(8))) float v8f;

#define NEG_INF (-9.0e15f)
#define ALPHA 0.2f
#define NB 4096
#define KC 128

// ---------------------------------------------------------------------------
// Kernel 1: Wh = h @ W  (row-major, 32768 x 64), s1 = Wh@a1, s2 = Wh@a2
// Block: 256 threads = 4 rows x 64 cols. W staged in LDS.
// ---------------------------------------------------------------------------
__global__ void gat_wh_kernel(const float* __restrict__ h,
                              const float* __restrict__ W,
                              const float* __restrict__ a,
                              float* __restrict__ Wh,
                              float* __restrict__ s1,
                              float* __restrict__ s2) {
    __shared__ float Wlds[64 * 64];
    __shared__ float hLds[4][64];
    __shared__ float redA[256];
    __shared__ float redB[256];

    const int tid = threadIdx.x;
    for (int idx = tid; idx < 64 * 64; idx += 256) Wlds[idx] = W[idx];

    const int r = tid >> 6;
    const int o = tid & 63;
    const size_t row = (size_t)blockIdx.x * 4 + r;

    hLds[r][o] = h[row * 64 + o];
    __syncthreads();

    float acc = 0.f;
#pragma unroll
    for (int k = 0; k < 64; ++k) acc += hLds[r][k] * Wlds[k * 64 + o];

    Wh[row * 64 + o] = acc;
    redA[tid] = acc * a[o];        // a1 = a[0:64]
    redB[tid] = acc * a[64 + o];   // a2 = a[64:128]
    __syncthreads();

#pragma unroll
    for (int st = 32; st > 0; st >>= 1) {
        if (o < st) {
            redA[tid] += redA[tid + st];
            redB[tid] += redB[tid + st];
        }
        __syncthreads();
    }
    if (o == 0) { s1[row] = redA[tid]; s2[row] = redB[tid]; }
}

// ---------------------------------------------------------------------------
// Kernel 2: fused masked softmax + att output + h' = att @ Wh (WMMA f32) + ELU
// One block per (batch, 16-row tile). 128 threads = 4 waves.
// Wave w owns output columns [16w, 16w+16); fp32 acc lives in VGPRs.
// ---------------------------------------------------------------------------
__global__ void gat_attn_kernel(const int* __restrict__ adj,
                                const float* __restrict__ Wh,
                                const float* __restrict__ s1g,
                                const float* __restrict__ s2g,
                                float* __restrict__ out_h,
                                float* __restrict__ out_att) {
    __shared__ float whLds[KC * 64];       // 32 KB: Wh[j0:j0+128][0:64]
    __shared__ float attLds[16 * 129];     // 8 KB: padded att tile
    __shared__ float redM[128];
    __shared__ float redS[128];
    __shared__ float rowM[16];
    __shared__ float rowInvS[16];
    __shared__ float rowS1[16];

    const int tid = threadIdx.x;
    const int b  = blockIdx.x >> 8;        // 256 row-tiles per batch
    const int i0 = (blockIdx.x & 255) * 16;
    const size_t rowBase = (size_t)b * NB;             // into Wh / s1 / s2 rows
    const size_t adjBase = (size_t)b * NB * NB;

    if (tid < 16) rowS1[tid] = s1g[rowBase + i0 + tid];
    __syncthreads();

    // ---- Phase 1: online (max, sum-of-exp) per row over all j --------------
    for (int r = 0; r < 16; ++r) {
        const float s1v = rowS1[r];
        const int* adjRow = adj + adjBase + (size_t)(i0 + r) * NB;
        float m = -3.0e38f, s = 0.f;
        for (int j = tid; j < NB; j += 128) {
            float e = s1v + s2g[rowBase + j];
            e = e > 0.f ? e : ALPHA * e;
            float v = adjRow[j] > 0 ? e : NEG_INF;
            if (v > m) { s = s * __expf(m - v) + 1.0f; m = v; }
            else       { s += __expf(v - m); }
        }
        redM[tid] = m; redS[tid] = s;
        __syncthreads();
#pragma unroll
        for (int st = 64; st > 0; st >>= 1) {
            if (tid < st) {
                float m1 = redM[tid],      sA = redS[tid];
                float m2 = redM[tid + st], sB = redS[tid + st];
                float mm = m1 > m2 ? m1 : m2;
                redM[tid] = mm;
                redS[tid] = sA * __expf(m1 - mm) + sB * __expf(m2 - mm);
            }
            __syncthreads();
        }
        if (tid == 0) { rowM[r] = redM[0]; rowInvS[r] = 1.0f / redS[0]; }
        __syncthreads();
    }

    // ---- Phase 2: per 128-col chunk: att write + WMMA accumulate -----------
    const int wave  = tid >> 5;
    const int lane  = tid & 31;
    const int half  = lane >> 4;   // 0: K0/K1 half, 1: K2/K3 half
    const int lp    = lane & 15;
    const int nbase = wave * 16;

    v8f acc = {};   // 16x16 fp32 accumulator tile per wave

    for (int j0 = 0; j0 < NB; j0 += KC) {
        // stage Wh chunk (L2-resident: Wh[b] is 1 MB, all batches fit in L2)
        const float* whSrc = Wh + (rowBase + j0) * 64;
        for (int idx = tid; idx < KC * 64; idx += 128) whLds[idx] = whSrc[idx];

        // normalized attention for 16 rows x 128 cols; also the graph output
        const int j = j0 + tid;
        const float s2v = s2g[rowBase + j];
#pragma unroll 4
        for (int r = 0; r < 16; ++r) {
            float e = rowS1[r] + s2v;
            e = e > 0.f ? e : ALPHA * e;
            const size_t aidx = adjBase + (size_t)(i0 + r) * NB + j;
            float v = adj[aidx] > 0 ? e : NEG_INF;
            float att = __expf(v - rowM[r]) * rowInvS[r];
            attLds[r * 129 + tid] = att;
            out_att[aidx] = att;
        }
        __syncthreads();

        // K-loop: V_WMMA_F32_16X16X4_F32, fp32-exact matrix math
#pragma unroll 8
        for (int kk = 0; kk < KC; kk += 4) {
            const int k0 = kk + 2 * half;
            v2f A, B;
            A[0] = attLds[lp * 129 + k0];            // A: M=lp, K=k0
            A[1] = attLds[lp * 129 + k0 + 1];        //    M=lp, K=k0+1
            B[0] = whLds[k0 * 64 + nbase + lp];      // B: K=k0,   N=lp
            B[1] = whLds[(k0 + 1) * 64 + nbase + lp];
            acc = __builtin_amdgcn_wmma_f32_16x16x4_f32(
                false, A, false, B, (short)0, acc, false, false);
        }
        __syncthreads();
    }

    // ---- Epilogue: ELU directly on the accumulator (C/D VGPR layout) -------
#pragma unroll
    for (int r = 0; r < 8; ++r) {
        const int gi = i0 + r + 8 * half;            // VGPR r: M=r (lo half) / r+8
        float x = acc[r];
        float y = x > 0.f ? x : (__expf(x) - 1.0f);
        out_h[(rowBase + gi) * 64 + nbase + lp] = y;
    }
}

// ---------------------------------------------------------------------------
extern "C" void kernel_launch(void* const* d_in, const int* in_sizes, int n_in,
                              void* d_out, int out_size, void* d_ws, size_t ws_size,
                              hipStream_t stream) {
    const float* h   = (const float*)d_in[0];
    const int*   adj = (const int*)d_in[1];
    const float* W   = (const float*)d_in[2];
    const float* a   = (const float*)d_in[3];

    float* out     = (float*)d_out;
    float* out_h   = out;                              // elu(h'): 8*4096*64
    float* out_att = out + (size_t)8 * NB * 64;        // att: 8*4096*4096

    float* Wh = (float*)d_ws;                          // 8*4096*64 floats (8 MB)
    float* s1 = Wh + (size_t)8 * NB * 64;              // 32768 floats
    float* s2 = s1 + (size_t)8 * NB;                   // 32768 floats

    gat_wh_kernel<<<8192, 256, 0, stream>>>(h, W, a, Wh, s1, s2);
    gat_attn_kernel<<<2048, 128, 0, stream>>>(adj, Wh, s1, s2, out_h, out_att);
}